// HANConv_61830349193884
// MI455X (gfx1250) — compile-verified
//
#include <hip/hip_runtime.h>
#include <math.h>

// ---------------------------------------------------------------------------
// HANConv for MI455X (gfx1250, wave32).
//
// Fusion: h = x@W_fc + b_fc is only consumed via h@Ws, h@Wd and the attention
// contractions, so we precompute fused weights Wf[256 x 160] per node type:
//   cols [  0, 64): xs for 1st src-role edge type
//   cols [ 64,128): xs for 2nd src-role edge type
//   cols [128,136): alpha_src for 1st src-role (att contraction folded in)
//   cols [136,144): alpha_src for 2nd src-role
//   cols [144,152): alpha_dst for 1st dst-role
//   cols [152,160): alpha_dst for 2nd dst-role
// One fp32 WMMA GEMM per node type produces XP = X @ Wf + bf.
// ---------------------------------------------------------------------------

typedef __attribute__((ext_vector_type(2))) float v2f;
typedef __attribute__((ext_vector_type(8))) float v8f;

#define IN_DIM 256
#define OC     64
#define HH     8
#define LDXP   160
#define SLOPE  0.2f
#define NT_COL 10        // 10 column tiles of 16 = 160 cols per wave

// order-preserving float <-> uint mapping (for integer atomicMax segment-max)
__device__ __forceinline__ unsigned encf(float f) {
    int i = __float_as_int(f);
    return (i >= 0) ? ((unsigned)i | 0x80000000u) : ~(unsigned)i;
}
__device__ __forceinline__ float decf(unsigned u) {
    int i = (u & 0x80000000u) ? (int)(u & 0x7fffffffu) : (int)(~u);
    return __int_as_float(i);
}
#define ENC_NEG_INF 0x007FFFFFu   // encf(-inf)

__device__ __forceinline__ float lrelu(float x) { return x > 0.f ? x : SLOPE * x; }

// ---------------------------------------------------------------------------
// Fused weight build: Wf[256 x 160], bf[160] per node type.
// grid (160, 2), block 256.
// ---------------------------------------------------------------------------
__global__ void prep_weights(const float* __restrict__ Wfc_a, const float* __restrict__ bfc_a,
                             const float* __restrict__ Wfc_p, const float* __restrict__ bfc_p,
                             const float* __restrict__ Wsrc,  const float* __restrict__ Wdst,
                             const float* __restrict__ att_s, const float* __restrict__ att_d,
                             float* __restrict__ Wf_a, float* __restrict__ bf_a,
                             float* __restrict__ Wf_p, float* __restrict__ bf_p)
{
    const int j = blockIdx.x;          // 0..159
    const int T = blockIdx.y;          // 0 = author, 1 = paper
    const int srcA = (T == 0) ? 0 : 1;
    const int srcB = (T == 0) ? 2 : 3;
    const int dstC = (T == 0) ? 1 : 0;
    const int dstD = (T == 0) ? 2 : 3;
    const float* Wfc = (T == 0) ? Wfc_a : Wfc_p;
    const float* bfc = (T == 0) ? bfc_a : bfc_p;
    float* Wf = (T == 0) ? Wf_a : Wf_p;
    float* bf = (T == 0) ? bf_a : bf_p;

    float wcat[8];
#pragma unroll
    for (int k = 0; k < 8; ++k) {
        float v;
        if (j < 64) {
            v = Wsrc[(srcA * 8 + k) * 64 + j];
        } else if (j < 128) {
            v = Wsrc[(srcB * 8 + k) * 64 + (j - 64)];
        } else if (j < 136) {
            int h = j - 128; v = 0.f;
            for (int c = 0; c < 8; ++c)
                v += Wsrc[(srcA * 8 + k) * 64 + h * 8 + c] * att_s[srcA * 64 + h * 8 + c];
        } else if (j < 144) {
            int h = j - 136; v = 0.f;
            for (int c = 0; c < 8; ++c)
                v += Wsrc[(srcB * 8 + k) * 64 + h * 8 + c] * att_s[srcB * 64 + h * 8 + c];
        } else if (j < 152) {
            int h = j - 144; v = 0.f;
            for (int c = 0; c < 8; ++c)
                v += Wdst[(dstC * 8 + k) * 64 + h * 8 + c] * att_d[dstC * 64 + h * 8 + c];
        } else {
            int h = j - 152; v = 0.f;
            for (int c = 0; c < 8; ++c)
                v += Wdst[(dstD * 8 + k) * 64 + h * 8 + c] * att_d[dstD * 64 + h * 8 + c];
        }
        wcat[k] = v;
    }

    const int i = threadIdx.x;          // 0..255 : K row of fused weight
    float s = 0.f;
#pragma unroll
    for (int k = 0; k < 8; ++k) s += Wfc[i * 8 + k] * wcat[k];
    Wf[(size_t)i * LDXP + j] = s;
    if (i == 0) {
        float b = 0.f;
#pragma unroll
        for (int k = 0; k < 8; ++k) b += bfc[k] * wcat[k];
        bf[j] = b;
    }
}

// ---------------------------------------------------------------------------
// Main GEMM: XP[N x 160] = X[N x 256] @ Wf[256 x 160] + bf, fp32 WMMA.
// One wave computes a 16-row x 160-col stripe (10 accumulators of 16x16).
// block = 256 (8 waves); assumes N multiple of 16 (100000 = 6250*16).
// ---------------------------------------------------------------------------
__global__ void __launch_bounds__(256)
gemm_wmma(const float* __restrict__ X, const float* __restrict__ Wf,
          const float* __restrict__ bf, float* __restrict__ XP, int nrows)
{
    const int wave = threadIdx.x >> 5;
    const int lane = threadIdx.x & 31;
    const int row0 = (blockIdx.x * 8 + wave) * 16;
    if (row0 >= nrows) return;                 // wave-uniform: EXEC stays all-1s

    const int rlo = lane & 15;                 // row within A-tile / col within B-tile
    const int khi = lane >> 4;                 // K-pair selector (0 -> K0,K1 ; 1 -> K2,K3)

    v8f acc[NT_COL];
#pragma unroll
    for (int t = 0; t < NT_COL; ++t)
#pragma unroll
        for (int v = 0; v < 8; ++v) acc[t][v] = 0.f;

    const float* xrow = X + (size_t)(row0 + rlo) * IN_DIM;
    for (int k0 = 0; k0 < IN_DIM; k0 += 4) {
        const int k = k0 + khi * 2;
        // A 16x4 fp32: VGPR0 = K (k), VGPR1 = K (k+1); 8-byte aligned load
        const v2f a = *(const v2f*)(xrow + k);
#pragma unroll
        for (int t = 0; t < NT_COL; ++t) {
            v2f b;
            b.x = Wf[(size_t)k * LDXP + t * 16 + rlo];
            b.y = Wf[(size_t)(k + 1) * LDXP + t * 16 + rlo];
            acc[t] = __builtin_amdgcn_wmma_f32_16x16x4_f32(
                false, a, false, b, (short)0, acc[t], false, false);
        }
    }

    // D layout: VGPR v holds M=v (lanes 0-15) / M=v+8 (lanes 16-31), N=lane&15
    const int rbase = row0 + khi * 8;
#pragma unroll
    for (int t = 0; t < NT_COL; ++t) {
        const int col = t * 16 + rlo;
        const float bb = bf[col];
#pragma unroll
        for (int v = 0; v < 8; ++v)
            XP[(size_t)(rbase + v) * LDXP + col] = acc[t][v] + bb;
    }
}

// ---------------------------------------------------------------------------
// Initialization kernels (workspace is poisoned; must rebuild every call)
// ---------------------------------------------------------------------------
__global__ void init_outs(float* __restrict__ m0a, float* __restrict__ m0p,
                          float* __restrict__ m1a, float* __restrict__ m1p,
                          const float* __restrict__ gbias, float* __restrict__ key_acc,
                          int N)
{
    const size_t idx = (size_t)blockIdx.x * blockDim.x + threadIdx.x;
    if (blockIdx.x == 0 && threadIdx.x < 256) key_acc[threadIdx.x] = 0.f;
    const size_t NOC = (size_t)N * OC;
    if (idx >= 4 * NOC) return;
    const int mat = (int)(idx / NOC);
    const size_t r = idx % NOC;
    const int j = (int)(r % OC);
    // out_a_pa uses gat_bias[1]; out_p_ap -> [0]; out_aa -> [2]; out_pp -> [3]
    if      (mat == 0) m0a[r] = gbias[1 * OC + j];
    else if (mat == 1) m0p[r] = gbias[0 * OC + j];
    else if (mat == 2) m1a[r] = gbias[2 * OC + j];
    else               m1p[r] = gbias[3 * OC + j];
}

__global__ void init_mz(unsigned* __restrict__ menc, float* __restrict__ z, int n)
{
    const int i = blockIdx.x * blockDim.x + threadIdx.x;
    if (i < n) { menc[i] = ENC_NEG_INF; z[i] = 0.f; }
}

// ---------------------------------------------------------------------------
// Edge passes (per edge type). alpha arrays live inside XP (row stride 160).
// ---------------------------------------------------------------------------
__global__ void edge_max(const int* __restrict__ ei, const float* __restrict__ as,
                         const float* __restrict__ ad, unsigned* __restrict__ menc, int E)
{
    const int gid = blockIdx.x * blockDim.x + threadIdx.x;
    if (gid >= E * HH) return;
    const int e = gid >> 3, h = gid & 7;
    const int s = ei[e], d = ei[E + e];
    const float x = lrelu(as[(size_t)s * LDXP + h] + ad[(size_t)d * LDXP + h]);
    atomicMax(&menc[(size_t)d * HH + h], encf(x));
}

__global__ void edge_expsum(const int* __restrict__ ei, const float* __restrict__ as,
                            const float* __restrict__ ad, const unsigned* __restrict__ menc,
                            float* __restrict__ z, int E)
{
    const int gid = blockIdx.x * blockDim.x + threadIdx.x;
    if (gid >= E * HH) return;
    const int e = gid >> 3, h = gid & 7;
    const int s = ei[e], d = ei[E + e];
    const float x = lrelu(as[(size_t)s * LDXP + h] + ad[(size_t)d * LDXP + h]);
    const float m = decf(menc[(size_t)d * HH + h]);
    atomicAdd(&z[(size_t)d * HH + h], expf(x - m));
}

__global__ void edge_scatter(const int* __restrict__ ei, const float* __restrict__ as,
                             const float* __restrict__ ad, const float* __restrict__ xs,
                             const unsigned* __restrict__ menc, const float* __restrict__ z,
                             float* __restrict__ outp, int E)
{
    const long long gid = (long long)blockIdx.x * blockDim.x + threadIdx.x;
    if (gid >= (long long)E * OC) return;
    const int e = (int)(gid >> 6);
    const int j = (int)(gid & 63);
    const int h = j >> 3;
    const int s = ei[e], d = ei[E + e];
    const float x = lrelu(as[(size_t)s * LDXP + h] + ad[(size_t)d * LDXP + h]);
    const float m = decf(menc[(size_t)d * HH + h]);
    const float w = expf(x - m);
    const float alpha = w / (z[(size_t)d * HH + h] + 1e-16f);
    atomicAdd(&outp[(size_t)d * OC + j], xs[(size_t)s * LDXP + j] * alpha);
}

// ---------------------------------------------------------------------------
// Semantic attention
// ---------------------------------------------------------------------------
__global__ void sem_key(const float* __restrict__ m0a, const float* __restrict__ m1a,
                        const float* __restrict__ m0p, const float* __restrict__ m1p,
                        const float* __restrict__ W_k, const float* __restrict__ b_k,
                        float* __restrict__ key_acc, int N)
{
    __shared__ float Wl[64 * 64];
    __shared__ float red[256];
    const int tid = threadIdx.x;
    for (int i = tid; i < 4096; i += 256) Wl[i] = W_k[i];
    __syncthreads();

    const int mi = blockIdx.y;           // 0: a/pa  1: a/aa  2: p/ap  3: p/pp
    const float* M = (mi == 0) ? m0a : (mi == 1) ? m1a : (mi == 2) ? m0p : m1p;
    const int col = tid & 63, sub = tid >> 6;
    const float bc = b_k[col];
    float part = 0.f;
    for (int n = blockIdx.x * 4 + sub; n < N; n += gridDim.x * 4) {
        const float* row = M + (size_t)n * OC;
        float t = bc;
#pragma unroll
        for (int k = 0; k < 64; ++k) t += row[k] * Wl[k * 64 + col];
        part += tanhf(t);
    }
    red[tid] = part;
    __syncthreads();
    if (tid < 64) {
        const float s = red[tid] + red[tid + 64] + red[tid + 128] + red[tid + 192];
        atomicAdd(&key_acc[mi * 64 + tid], s);
    }
}

__global__ void sem_score(const float* __restrict__ key_acc, const float* __restrict__ q,
                          float* __restrict__ scores, int N)
{
    if (threadIdx.x == 0 && blockIdx.x == 0) {
        float s[4];
        for (int mi = 0; mi < 4; ++mi) {
            float acc = 0.f;
            for (int j = 0; j < 64; ++j) acc += q[j] * key_acc[mi * 64 + j];
            s[mi] = acc / (float)N;
        }
        {   // author: (pa, aa)
            const float mx = fmaxf(s[0], s[1]);
            const float e0 = expf(s[0] - mx), e1 = expf(s[1] - mx);
            scores[0] = e0 / (e0 + e1); scores[1] = e1 / (e0 + e1);
        }
        {   // paper: (ap, pp)
            const float mx = fmaxf(s[2], s[3]);
            const float e0 = expf(s[2] - mx), e1 = expf(s[3] - mx);
            scores[2] = e0 / (e0 + e1); scores[3] = e1 / (e0 + e1);
        }
    }
}

__global__ void sem_combine(float* __restrict__ out, const float* __restrict__ m1a,
                            const float* __restrict__ m1p, const float* __restrict__ scores,
                            size_t NOC)
{
    const size_t idx = (size_t)blockIdx.x * blockDim.x + threadIdx.x;
    if (idx >= 2 * NOC) return;
    if (idx < NOC) out[idx] = scores[0] * out[idx] + scores[1] * m1a[idx];
    else {
        const size_t r = idx - NOC;
        out[idx] = scores[2] * out[idx] + scores[3] * m1p[r];
    }
}

// ---------------------------------------------------------------------------
// Launch
// ---------------------------------------------------------------------------
extern "C" void kernel_launch(void* const* d_in, const int* in_sizes, int n_in,
                              void* d_out, int out_size, void* d_ws, size_t ws_size,
                              hipStream_t stream)
{
    const float* x_a   = (const float*)d_in[0];
    const float* x_p   = (const float*)d_in[1];
    const int*   ei[4] = { (const int*)d_in[2], (const int*)d_in[3],
                           (const int*)d_in[4], (const int*)d_in[5] };
    const float* Wfc_a = (const float*)d_in[6];
    const float* bfc_a = (const float*)d_in[7];
    const float* Wfc_p = (const float*)d_in[8];
    const float* bfc_p = (const float*)d_in[9];
    const float* Wsrc  = (const float*)d_in[10];
    const float* Wdst  = (const float*)d_in[11];
    const float* atts  = (const float*)d_in[12];
    const float* attd  = (const float*)d_in[13];
    const float* gbias = (const float*)d_in[14];
    const float* W_k   = (const float*)d_in[15];
    const float* b_k   = (const float*)d_in[16];
    const float* q     = (const float*)d_in[17];

    const int N = in_sizes[0] / IN_DIM;   // 100000
    const int E = in_sizes[2] / 2;        // 1600000

    // workspace carve-up (floats)
    float* ws = (float*)d_ws;
    size_t off = 0;
    float* Wf_a = ws + off; off += 256 * LDXP;
    float* Wf_p = ws + off; off += 256 * LDXP;
    float* bf_a = ws + off; off += LDXP;
    float* bf_p = ws + off; off += LDXP;
    float* XP_a = ws + off; off += (size_t)N * LDXP;
    float* XP_p = ws + off; off += (size_t)N * LDXP;
    float* out_aa = ws + off; off += (size_t)N * OC;   // author m1 (a->a)
    float* out_pp = ws + off; off += (size_t)N * OC;   // paper  m1 (p->p)
    unsigned* menc = (unsigned*)(ws + off); off += (size_t)N * HH;
    float* zbuf = ws + off; off += (size_t)N * HH;
    float* key_acc = ws + off; off += 4 * OC;
    float* scores = ws + off; off += 4;

    float* outf = (float*)d_out;
    float* out_a_pa = outf;                     // author m0
    float* out_p_ap = outf + (size_t)N * OC;    // paper  m0

    // 1) fused weights
    prep_weights<<<dim3(160, 2), 256, 0, stream>>>(Wfc_a, bfc_a, Wfc_p, bfc_p,
                                                   Wsrc, Wdst, atts, attd,
                                                   Wf_a, bf_a, Wf_p, bf_p);

    // 2) init output accumulators (GAT bias folded in) + semantic key acc
    {
        const size_t tot = 4 * (size_t)N * OC;
        init_outs<<<(int)((tot + 255) / 256), 256, 0, stream>>>(
            out_a_pa, out_p_ap, out_aa, out_pp, gbias, key_acc, N);
    }

    // 3) fused projection GEMMs (WMMA fp32)
    {
        const int ntiles = N / 16;
        const int gblocks = (ntiles + 7) / 8;
        gemm_wmma<<<gblocks, 256, 0, stream>>>(x_a, Wf_a, bf_a, XP_a, N);
        gemm_wmma<<<gblocks, 256, 0, stream>>>(x_p, Wf_p, bf_p, XP_p, N);
    }

    // 4) per-edge-type GAT passes
    const float* as_t[4]  = { XP_a + 128, XP_p + 128, XP_a + 136, XP_p + 136 };
    const float* ad_t[4]  = { XP_p + 144, XP_a + 144, XP_a + 152, XP_p + 152 };
    const float* xs_t[4]  = { XP_a,       XP_p,       XP_a + 64,  XP_p + 64  };
    float*       out_t[4] = { out_p_ap,   out_a_pa,   out_aa,     out_pp     };

    const int mzBlocks = (N * HH + 255) / 256;
    const int ehBlocks = (E * HH + 255) / 256;
    const int esBlocks = (int)(((long long)E * OC + 255) / 256);
    for (int t = 0; t < 4; ++t) {
        init_mz<<<mzBlocks, 256, 0, stream>>>(menc, zbuf, N * HH);
        edge_max<<<ehBlocks, 256, 0, stream>>>(ei[t], as_t[t], ad_t[t], menc, E);
        edge_expsum<<<ehBlocks, 256, 0, stream>>>(ei[t], as_t[t], ad_t[t], menc, zbuf, E);
        edge_scatter<<<esBlocks, 256, 0, stream>>>(ei[t], as_t[t], ad_t[t], xs_t[t],
                                                   menc, zbuf, out_t[t], E);
    }

    // 5) semantic attention
    sem_key<<<dim3(1024, 4), 256, 0, stream>>>(out_a_pa, out_aa, out_p_ap, out_pp,
                                               W_k, b_k, key_acc, N);
    sem_score<<<1, 64, 0, stream>>>(key_acc, q, scores, N);
    {
        const size_t NOC = (size_t)N * OC;
        sem_combine<<<(int)((2 * NOC + 255) / 256), 256, 0, stream>>>(
            outf, out_aa, out_pp, scores, NOC);
    }
}